// TwoLayerGraphSAGE_47691316854881
// MI455X (gfx1250) — compile-verified
//
#include <hip/hip_runtime.h>
#include <cstdint>
#include <cstddef>

typedef float v2f __attribute__((ext_vector_type(2)));
typedef float v8f __attribute__((ext_vector_type(8)));

#define N_NODES 100000
#define N_EDGES 1600000
#define D_IN    128
#define D_HID   128
#define D_OUT   64
#define KDIM    128           // contraction dim, both layers
#define ROWS_PB 32            // node rows per GEMM block

// ---------------------------------------------------------------------------
// Zero a float buffer (workspace is poisoned between replays; re-zero per call)
// ---------------------------------------------------------------------------
__global__ void zero_f32(float* __restrict__ p, int n) {
  int i = blockIdx.x * blockDim.x + threadIdx.x;
  if (i < n) p[i] = 0.0f;
}

// ---------------------------------------------------------------------------
// Degree accumulation: deg[dst] += 1 per edge
// ---------------------------------------------------------------------------
__global__ void degree_kernel(const long long* __restrict__ ei,
                              float* __restrict__ deg, int E) {
  int e = blockIdx.x * blockDim.x + threadIdx.x;
  if (e < E) {
    int dst = (int)ei[(size_t)E + e];
    __hip_atomic_fetch_add(&deg[dst], 1.0f,
                           __ATOMIC_RELAXED, __HIP_MEMORY_SCOPE_AGENT);
  }
}

// ---------------------------------------------------------------------------
// invdeg: deg[n] = 1 / max(deg[n], 1)   (mean-divide folded into A staging)
// ---------------------------------------------------------------------------
__global__ void invdeg_kernel(float* __restrict__ deg, int n) {
  int i = blockIdx.x * blockDim.x + threadIdx.x;
  if (i < n) deg[i] = 1.0f / fmaxf(deg[i], 1.0f);
}

// ---------------------------------------------------------------------------
// Edge scatter: one wave (32 lanes) per edge; each lane carries a float4.
// Node table (51.2 MB) is L2-resident (192 MB L2); L2 atomic units reduce.
// ---------------------------------------------------------------------------
__global__ void scatter_kernel(const float* __restrict__ feat,
                               float* __restrict__ agg,
                               const long long* __restrict__ ei, int E) {
  int t = blockIdx.x * blockDim.x + threadIdx.x;
  int e = t >> 5;
  if (e >= E) return;
  int lane = t & 31;
  int src = (int)ei[e];
  int dst = (int)ei[(size_t)E + e];
  const float4 v = ((const float4*)(feat + (size_t)src * D_IN))[lane];
  float* o = agg + (size_t)dst * D_IN + lane * 4;
  __hip_atomic_fetch_add(o + 0, v.x, __ATOMIC_RELAXED, __HIP_MEMORY_SCOPE_AGENT);
  __hip_atomic_fetch_add(o + 1, v.y, __ATOMIC_RELAXED, __HIP_MEMORY_SCOPE_AGENT);
  __hip_atomic_fetch_add(o + 2, v.z, __ATOMIC_RELAXED, __HIP_MEMORY_SCOPE_AGENT);
  __hip_atomic_fetch_add(o + 3, v.w, __ATOMIC_RELAXED, __HIP_MEMORY_SCOPE_AGENT);
}

// ---------------------------------------------------------------------------
// f32 WMMA wrapper: D = A(16x4) * B(4x16) + C, full-precision matrix pipe
// ---------------------------------------------------------------------------
__device__ __forceinline__ v8f wmma_f32(v2f a, v2f b, v8f c) {
  return __builtin_amdgcn_wmma_f32_16x16x4_f32(
      /*neg_a=*/false, a, /*neg_b=*/false, b,
      /*c_mod=*/(short)0, c, /*reuse_a=*/false, /*reuse_b=*/false);
}

// Generic shared pointers carry the LDS byte offset in their low 32 bits
// (flat->LDS aperture).
__device__ __forceinline__ unsigned lds_byte_addr(const void* p) {
  return (unsigned)(uintptr_t)p;
}

// Async DMA: copy 16 bytes from global to LDS per active lane (ASYNCcnt).
__device__ __forceinline__ void async_copy_b128(unsigned lds_addr,
                                                const void* gaddr) {
  asm volatile("global_load_async_to_lds_b128 %0, %1, off"
               :: "v"(lds_addr), "v"(gaddr)
               : "memory");
}

__device__ __forceinline__ void wait_asynccnt0() {
#if __has_builtin(__builtin_amdgcn_s_wait_asynccnt)
  __builtin_amdgcn_s_wait_asynccnt(0);
#else
  asm volatile("s_wait_asynccnt 0" ::: "memory");
#endif
}

// ---------------------------------------------------------------------------
// Fused SAGEConv GEMM:  out = act( (agg*invdeg) @ Wl + bias + self @ Wr )
//
// Block tile: ROWS_PB(32) rows x N cols, (N/32)*2 waves. Staging:
//   - self tile  : GLOBAL_LOAD_ASYNC_TO_LDS_B128 (TDM-style async DMA,
//                  overlapped with the manual staging below)
//   - weights    : packed in B-fragment pair order — float2
//                  (W[2k2][c], W[2k2+1][c]) at pair index [k2*N + c]
//   - agg tile   : manual copy, scaled by invdeg (mean aggregate)
// N is a template constant and the K loop is fully unrolled, so every LDS
// access in the hot loop is base + immediate offset: ds_load_b64 + v_wmma.
// ---------------------------------------------------------------------------
template <int N, bool RELU>
__global__ void sage_gemm(const float* __restrict__ agg,
                          const float* __restrict__ invdeg,
                          const float* __restrict__ self,
                          const float* __restrict__ Wl,
                          const float* __restrict__ Wr,
                          const float* __restrict__ bias,
                          float* __restrict__ out) {
  extern __shared__ float smem[];
  float* ldsW = smem;                            // 2*KDIM*N floats (packed pairs)
  float* ldsA = ldsW + 2 * KDIM * N;             // ROWS_PB*KDIM (agg, scaled)
  float* ldsX = ldsA + ROWS_PB * KDIM;           // ROWS_PB*KDIM (self)

  const int tid     = threadIdx.x;
  constexpr int colWaves = N >> 5;               // 4 (N=128) or 2 (N=64)
  constexpr int nthread  = colWaves * 2 * 32;    // 256 or 128
  const int rowBase = blockIdx.x * ROWS_PB;

  // ---- Kick off async DMA of the self tile (ASYNCcnt) ----------------------
  {
    constexpr int f4PerTile = (ROWS_PB * KDIM) >> 2;   // 1024
    for (int j = tid; j < f4PerTile; j += nthread) {
      const int row = j >> 5;                          // KDIM/4 = 32 f4 per row
      const int c0  = (j & 31) << 2;
      async_copy_b128(lds_byte_addr(ldsX + row * KDIM + c0),
                      self + (size_t)(rowBase + row) * KDIM + c0);
    }
  }

  // ---- Stage weights (packed), overlapping the async DMA -------------------
  {
    constexpr int f4PerMat = (KDIM * N) >> 2;
    for (int j = tid; j < 2 * f4PerMat; j += nthread) {
      const int m  = j >= f4PerMat;                    // 0 = Wl, 1 = Wr
      const int jj = m ? j - f4PerMat : j;
      const int k  = jj / (N >> 2);
      const int c0 = (jj - k * (N >> 2)) << 2;
      const float4 v = *(const float4*)((m ? Wr : Wl) + (size_t)k * N + c0);
      float* base = ldsW + (size_t)m * KDIM * N + (k >> 1) * (2 * N) + (k & 1);
      base[2 * (c0 + 0)] = v.x;
      base[2 * (c0 + 1)] = v.y;
      base[2 * (c0 + 2)] = v.z;
      base[2 * (c0 + 3)] = v.w;
    }
  }

  // ---- Stage agg tile (scaled by invdeg) -----------------------------------
  {
    constexpr int f4PerTile = (ROWS_PB * KDIM) >> 2;   // 1024
    for (int j = tid; j < f4PerTile; j += nthread) {
      const int row = j >> 5;
      const int c0  = (j & 31) << 2;
      const int gr  = rowBase + row;
      float4 v = *(const float4*)(agg + (size_t)gr * KDIM + c0);
      const float s = invdeg[gr];
      v.x *= s; v.y *= s; v.z *= s; v.w *= s;
      *(float4*)(ldsA + row * KDIM + c0) = v;
    }
  }

  wait_asynccnt0();     // our async copies have landed in LDS
  __syncthreads();      // ... and are visible to all waves in the block

  // ---- WMMA main loop (fully unrolled, immediate DS offsets) ---------------
  const int w      = tid >> 5;
  const int lane   = tid & 31;
  const int lrow   = lane & 15;
  const int half   = lane >> 4;
  const int rowSub = (w / colWaves) * 16;        // 0 or 16 within block tile
  const int col0   = (w % colWaves) * 32;
  const int cA     = col0 + lrow;
  const int cB     = cA + 16;

  const float* aRow = ldsA + (rowSub + lrow) * KDIM;
  const float* xRow = ldsX + (rowSub + lrow) * KDIM;
  const float2* wl2 = (const float2*)ldsW;                       // pair view
  const float2* wr2 = (const float2*)(ldsW + (size_t)KDIM * N);

  v8f acc0 = {};
  v8f acc1 = {};

#pragma unroll
  for (int k = 0; k < KDIM; k += 4) {
    const int ka = k + 2 * half;           // A/B K-offset for this lane half
    const int k2 = (k >> 1) + half;        // packed pair row

    const float2 a1 = *(const float2*)(aRow + ka);
    const float2 a2 = *(const float2*)(xRow + ka);
    v2f A1; A1.x = a1.x; A1.y = a1.y;
    v2f A2; A2.x = a2.x; A2.y = a2.y;

    const float2 bl0 = wl2[k2 * N + cA];
    const float2 bl1 = wl2[k2 * N + cB];
    const float2 br0 = wr2[k2 * N + cA];
    const float2 br1 = wr2[k2 * N + cB];
    v2f Bl0; Bl0.x = bl0.x; Bl0.y = bl0.y;
    v2f Bl1; Bl1.x = bl1.x; Bl1.y = bl1.y;
    v2f Br0; Br0.x = br0.x; Br0.y = br0.y;
    v2f Br1; Br1.x = br1.x; Br1.y = br1.y;

    acc0 = wmma_f32(A1, Bl0, acc0);
    acc0 = wmma_f32(A2, Br0, acc0);
    acc1 = wmma_f32(A1, Bl1, acc1);
    acc1 = wmma_f32(A2, Br1, acc1);
  }

  // ---- Epilogue: bias (+relu), store ---------------------------------------
  const float bA = bias[cA];
  const float bB = bias[cB];
#pragma unroll
  for (int r = 0; r < 8; ++r) {
    const int orow = rowBase + rowSub + r + 8 * half;  // C/D: VGPR r -> r, r+8
    float v0 = acc0[r] + bA;
    float v1 = acc1[r] + bB;
    if (RELU) { v0 = fmaxf(v0, 0.0f); v1 = fmaxf(v1, 0.0f); }
    out[(size_t)orow * N + cA] = v0;
    out[(size_t)orow * N + cB] = v1;
  }
}

// ---------------------------------------------------------------------------
// Host-side launch
// ---------------------------------------------------------------------------
extern "C" void kernel_launch(void* const* d_in, const int* in_sizes, int n_in,
                              void* d_out, int out_size, void* d_ws, size_t ws_size,
                              hipStream_t stream) {
  const float*     x   = (const float*)d_in[0];
  const long long* ei  = (const long long*)d_in[1];   // [2, E] int64
  // d_in[2] = drpt (identity at inference)
  const float*     W1l = (const float*)d_in[3];
  const float*     W1r = (const float*)d_in[4];
  const float*     b1  = (const float*)d_in[5];
  const float*     W2l = (const float*)d_in[6];
  const float*     W2r = (const float*)d_in[7];
  const float*     b2  = (const float*)d_in[8];
  float* out = (float*)d_out;

  // Workspace layout: [agg (N*128)] [deg (N)] [h (N*128)]  (~103 MB)
  float* agg = (float*)d_ws;
  float* deg = agg + (size_t)N_NODES * D_HID;
  float* h   = deg + N_NODES;

  const int AGG_ELEMS = N_NODES * D_HID;                 // 12,800,000
  const unsigned scatterBlocks =
      (unsigned)(((long long)N_EDGES * 32 + 255) / 256); // 200,000
  const int gemmBlocks = N_NODES / ROWS_PB;              // 3125

  const size_t shmem1 =
      (size_t)(2 * KDIM * D_HID + 2 * ROWS_PB * KDIM) * sizeof(float); // 160 KB
  const size_t shmem2 =
      (size_t)(2 * KDIM * D_OUT + 2 * ROWS_PB * KDIM) * sizeof(float); //  96 KB

  // ---- Layer 1 ----
  {
    int n = AGG_ELEMS + N_NODES;                         // zero agg + deg
    zero_f32<<<(n + 255) / 256, 256, 0, stream>>>(agg, n);
  }
  degree_kernel<<<(N_EDGES + 255) / 256, 256, 0, stream>>>(ei, deg, N_EDGES);
  scatter_kernel<<<scatterBlocks, 256, 0, stream>>>(x, agg, ei, N_EDGES);
  invdeg_kernel<<<(N_NODES + 255) / 256, 256, 0, stream>>>(deg, N_NODES);
  sage_gemm<D_HID, true><<<gemmBlocks, (D_HID / 32) * 2 * 32, shmem1, stream>>>(
      agg, deg, x, W1l, W1r, b1, h);

  // ---- Layer 2 ----
  zero_f32<<<(AGG_ELEMS + 255) / 256, 256, 0, stream>>>(agg, AGG_ELEMS);
  scatter_kernel<<<scatterBlocks, 256, 0, stream>>>(h, agg, ei, N_EDGES);
  sage_gemm<D_OUT, false><<<gemmBlocks, (D_OUT / 32) * 2 * 32, shmem2, stream>>>(
      agg, deg, h, W2l, W2r, b2, out);
}